// SAE_46282567582162
// MI455X (gfx1250) — compile-verified
//
#include <hip/hip_runtime.h>

typedef float v2f __attribute__((ext_vector_type(2)));
typedef float v8f __attribute__((ext_vector_type(8)));

// out = x @ M^T + cc with M = A4*A3*A2*A1 rank-1: M[m][k] = u[m]*w[k].
// Transposed WMMA:  D[m][n] = sum_k (u_m w_k) * x[row][k] + cc_m, so lane n
// holds out[row0+n][0..6] row-major -> b128+b96 stores. Grid-stride loop over
// 16-row tiles amortizes the wave-invariant preamble (A-fragment, bias, w/u).
__global__ __launch_bounds__(256) void spmm_chain_rank1_wmma(
    const float* __restrict__ x,
    const float* __restrict__ v1, const float* __restrict__ b1,
    const float* __restrict__ v2, const float* __restrict__ b2,
    const float* __restrict__ v3, const float* __restrict__ b3,
    const float* __restrict__ v4, const float* __restrict__ b4,
    float* __restrict__ out, int nrows, int ntiles, int wstride)
{
    const int  lane  = (int)(threadIdx.x & 31u);
    const int  wave0 = (int)((blockIdx.x * blockDim.x + threadIdx.x) >> 5);
    const int  idx   = lane & 15;
    const bool lo    = lane < 16;
    const int  kb    = lo ? 0 : 2;               // K-base of this lane-half (16x4 f32 layout)

    // ---- wave-invariant rank-1 factors (scalar loads -> s_fmac_f32) ----
    const float w0 = v2[0]*v1[0], w1 = v2[0]*v1[1], w2 = v2[0]*v1[2];
    const float w3 = v2[1]*v1[3];
    const float w4 = v2[2]*v1[4], w5 = v2[2]*v1[5], w6 = v2[2]*v1[6];
    const float beta = b1[0]*v2[0] + b1[1]*v2[1] + b1[2]*v2[2] + b2[0];

    // ---- loop-invariant A-fragment: A[m][k] = u_m * w_k (rows m>=7 zero) ----
    const int   mc   = (idx < 7) ? idx : 6;
    const int   mapm = (mc > 2) + (mc > 3);      // column pattern {0,0,0,1,2,2,2}
    const float um   = (idx < 7) ? v4[mc] * v3[mapm] : 0.0f;
    v2f Af0, Af1;
    Af0.x = um * (lo ? w0 : w2);                 // K = kb
    Af0.y = um * (lo ? w1 : w3);                 // K = kb+1
    Af1.x = um * (lo ? w4 : w6);                 // K = kb+4
    Af1.y = lo ? um * w5 : 0.0f;                 // K = kb+5 (w7 = 0 pad)

    // ---- loop-invariant bias cc_m (held in SGPRs; re-moved into acc) ----
    const float uu0 = v4[0]*v3[0], uu1 = v4[1]*v3[0], uu2 = v4[2]*v3[0];
    const float uu3 = v4[3]*v3[1];
    const float uu4 = v4[4]*v3[2], uu5 = v4[5]*v3[2], uu6 = v4[6]*v3[2];
    const float cc0 = beta*uu0 + (v4[0]*b3[0] + b4[0]);
    const float cc1 = beta*uu1 + (v4[1]*b3[0] + b4[1]);
    const float cc2 = beta*uu2 + (v4[2]*b3[0] + b4[2]);
    const float cc3 = beta*uu3 + (v4[3]*b3[1] + b4[3]);
    const float cc4 = beta*uu4 + (v4[4]*b3[2] + b4[4]);
    const float cc5 = beta*uu5 + (v4[5]*b3[2] + b4[5]);
    const float cc6 = beta*uu6 + (v4[6]*b3[2] + b4[6]);

    // ---- loop-invariant per-lane column offsets (hi half reads in-bounds
    //      dummy xr[6] for the K=7 slot; zeroed by cndmask below) ----
    const int c01 = kb;                          // columns kb, kb+1
    const int c4  = kb + 4;                      // column kb+4
    const int c5  = lo ? kb + 5 : 6;             // column kb+5 (dummy for hi)

    for (int tile = wave0; tile < ntiles; tile += wstride) {
        // ---- B-fragment: B[k][n] = x[tile*16+n][k] (coalesced 448B window) ----
        int brow = tile * 16 + idx;
        if (brow >= nrows) brow = nrows - 1;     // clamp keeps EXEC full on tail
        const float* xr = x + (size_t)brow * 7;
        v2f Bf0, Bf1;
        Bf0.x = xr[c01 + 0];
        Bf0.y = xr[c01 + 1];
        Bf1.x = xr[c4];
        {
            const float t = xr[c5];
            Bf1.y = lo ? t : 0.0f;               // K=7 zero pad
        }

        // ---- C[m][n] = cc_m, then two chained WMMAs over K = 0..3 / 4..7 ----
        v8f acc;
        acc[0] = cc0; acc[1] = cc1; acc[2] = cc2; acc[3] = cc3;
        acc[4] = cc4; acc[5] = cc5; acc[6] = cc6; acc[7] = 0.0f;

        acc = __builtin_amdgcn_wmma_f32_16x16x4_f32(false, Af0, false, Bf0,
                                                    (short)0, acc, false, false);
        acc = __builtin_amdgcn_wmma_f32_16x16x4_f32(false, Af1, false, Bf1,
                                                    (short)0, acc, false, false);

        // ---- store: lane n (0-15) owns full row out[tile*16+n][0..6] ----
        const int srow = tile * 16 + idx;
        if (lo && srow < nrows) {                // one predicate, b128+b96 stores
            float* o = out + (size_t)srow * 7;
            o[0] = acc[0];
            o[1] = acc[1];
            o[2] = acc[2];
            o[3] = acc[3];
            o[4] = acc[4];
            o[5] = acc[5];
            o[6] = acc[6];
        }
    }
}

extern "C" void kernel_launch(void* const* d_in, const int* in_sizes, int n_in,
                              void* d_out, int out_size, void* d_ws, size_t ws_size,
                              hipStream_t stream) {
    const float* x  = (const float*)d_in[0];
    const float* v1 = (const float*)d_in[1];
    const float* b1 = (const float*)d_in[2];
    const float* v2 = (const float*)d_in[3];
    const float* b2 = (const float*)d_in[4];
    const float* v3 = (const float*)d_in[5];
    const float* b3 = (const float*)d_in[6];
    const float* v4 = (const float*)d_in[7];
    const float* b4 = (const float*)d_in[8];
    float* out = (float*)d_out;

    const int nrows = in_sizes[0] / 7;                   // 4,194,304
    const int ntiles = (nrows + 15) / 16;                // 16-row tiles
    const int TILES_PER_WAVE = 8;                        // amortize preamble
    const int waves_per_block = 256 / 32;                // wave32
    const int waves  = (ntiles + TILES_PER_WAVE - 1) / TILES_PER_WAVE;
    const int blocks = (waves + waves_per_block - 1) / waves_per_block;
    const int wstride = blocks * waves_per_block;        // grid-stride in tiles

    spmm_chain_rank1_wmma<<<blocks, 256, 0, stream>>>(
        x, v1, b1, v2, b2, v3, b3, v4, b4, out, nrows, ntiles, wstride);
}